// FreqEncoder_8744553415209
// MI455X (gfx1250) — compile-verified
//
#include <hip/hip_runtime.h>
#include <math.h>

// Problem constants (match reference)
#define N_POINTS       524288
#define X_DIM          3
#define NUM_FREQS      8
#define NUM_FEATS      8
#define RES            256
#define F_TOTAL        (NUM_FREQS * 2 * X_DIM)          // 48
#define OUT_COLS       (X_DIM + NUM_FEATS * F_TOTAL)    // 387
#define F_PER_BLK      (2 * X_DIM)                      // 6 table rows per scale group
#define SLICE_FLOATS   (F_PER_BLK * NUM_FEATS * RES)    // 12288 floats = 48 KB
#define THREADS        256
#define PTS_PER_BLOCK  2048
#define PTS_PER_THREAD (PTS_PER_BLOCK / THREADS)        // 8
#define HALF_PI_F      1.57079632679489661923f

__global__ __launch_bounds__(THREADS)
void freq_encode_kernel(const float* __restrict__ x,
                        const float* __restrict__ features,
                        float* __restrict__ out,
                        int n_points)
{
    // Raw copy of features[s*6 .. s*6+5][C][RES] (contiguous slice), 48 KB.
    __shared__ float smem[SLICE_FLOATS];

    const int tid = threadIdx.x;
    const int s   = blockIdx.y;                       // scale group 0..7
    const float* gsrc = features + (size_t)s * SLICE_FLOATS;

    // ---- Stage the 48 KB table slice into LDS ----
#if defined(__gfx1250__) && \
    __has_builtin(__builtin_amdgcn_global_load_async_to_lds_b128) && \
    __has_builtin(__builtin_amdgcn_s_wait_asynccnt)
    {
        typedef int v4i __attribute__((vector_size(16)));
        typedef v4i __attribute__((address_space(1)))* g_v4i;
        typedef v4i __attribute__((address_space(3)))* l_v4i;
        // 12288 floats = 3072 x 16B chunks; 12 chunks per thread, coalesced.
        #pragma unroll
        for (int it = 0; it < SLICE_FLOATS / 4 / THREADS; ++it) {
            const int chunk = it * THREADS + tid;
            __builtin_amdgcn_global_load_async_to_lds_b128(
                (g_v4i)(v4i*)(void*)(const void*)(gsrc + chunk * 4),
                (l_v4i)(v4i*)(void*)(&smem[chunk * 4]),
                /*offset=*/0, /*cpol=*/0);
        }
        __builtin_amdgcn_s_wait_asynccnt(0);
    }
#else
    for (int i = tid; i < SLICE_FLOATS; i += THREADS) smem[i] = gsrc[i];
#endif
    __syncthreads();

    // linspace(1, 256, 8) in fp32: 1 + s * (255/7)
    const float scale = 1.0f + (float)s * (255.0f / 7.0f);
    const int pbase = blockIdx.x * PTS_PER_BLOCK;

    for (int k = 0; k < PTS_PER_THREAD; ++k) {
        const int p = pbase + k * THREADS + tid;
        if (p >= n_points) continue;

        const float xv0 = x[3 * p + 0];
        const float xv1 = x[3 * p + 1];
        const float xv2 = x[3 * p + 2];
        float* __restrict__ orow = out + (size_t)p * OUT_COLS;

        if (s == 0) {               // passthrough coords written exactly once
            orow[0] = xv0; orow[1] = xv1; orow[2] = xv2;
        }

        const float xv[X_DIM] = {xv0, xv1, xv2};

        #pragma unroll
        for (int t = 0; t < 2; ++t) {
            #pragma unroll
            for (int d = 0; d < X_DIM; ++d) {
                const int fLocal = t * X_DIM + d;
                // latent = sin(x*scale) for t=0, sin(x*scale + pi/2) for t=1
                const float arg = xv[d] * scale + (t ? HALF_PI_F : 0.0f);
                const float l   = sinf(arg);

                // align_corners=True mapping into [0, 255]
                float pos = (l + 1.0f) * 127.5f;
                pos = fminf(fmaxf(pos, 0.0f), 255.0f);
                int i0 = (int)pos;               // floor (pos >= 0)
                if (i0 > RES - 2) i0 = RES - 2;  // fold i1-clamp: w=1 at pos==255
                const float w = pos - (float)i0;

                // LDS slice layout == global: [fLocal][c][r]
                const float* __restrict__ frow =
                    &smem[fLocal * (NUM_FEATS * RES) + i0];
                float* __restrict__ ocol = orow + X_DIM + s * 6 + t * 3 + d;

                #pragma unroll
                for (int c = 0; c < NUM_FEATS; ++c) {
                    const float f0 = frow[c * RES];       // tap i0
                    const float f1 = frow[c * RES + 1];   // tap i0+1 (adjacent)
                    ocol[c * F_TOTAL] = f0 + (f1 - f0) * w + l;
                }
            }
        }
    }
}

extern "C" void kernel_launch(void* const* d_in, const int* in_sizes, int n_in,
                              void* d_out, int out_size, void* d_ws, size_t ws_size,
                              hipStream_t stream)
{
    const float* x        = (const float*)d_in[0];
    const float* features = (const float*)d_in[1];
    float* out            = (float*)d_out;

    const int n_points = in_sizes[0] / X_DIM;   // 524288
    const int xblocks  = (n_points + PTS_PER_BLOCK - 1) / PTS_PER_BLOCK;

    dim3 grid(xblocks, NUM_FREQS);              // y = scale group (6 table rows each)
    freq_encode_kernel<<<grid, THREADS, 0, stream>>>(x, features, out, n_points);
}